// SpatialFeatureMachine_77309411573
// MI455X (gfx1250) — compile-verified
//
#include <hip/hip_runtime.h>
#include <hip/hip_bf16.h>

// Problem constants from the reference
#define Bb 8
#define Tt 12
#define Nn 20000
#define Ee 320000
#define Ff 64

typedef __attribute__((ext_vector_type(2))) float v2f;
typedef __attribute__((ext_vector_type(8))) float v8f;

// ---------------------------------------------------------------------------
// Kernel 1: h[b,n,f] = sum_t x[b,t,n] * w_gcn[t,f]
// One wave per (b, 16-node tile). K=12 -> 3 chunks of 4 on V_WMMA_F32_16X16X4_F32.
// A 16x4 layout: lanes 0-15 hold M=0..15 with K = k0+{0,1} in {.x,.y};
//                lanes 16-31 hold K = k0+{2,3}. B 4x16 mirrored.
// D 16x16 f32: vgpr r, lanes 0-15 -> M=r, lanes 16-31 -> M=r+8, N=lane&15.
// ---------------------------------------------------------------------------
__global__ __launch_bounds__(256) void gcn_gemm1_wmma(const float* __restrict__ x,
                                                      const float* __restrict__ w_gcn,
                                                      float* __restrict__ h) {
  const int lane = threadIdx.x & 31;
  const int wave = blockIdx.x * 8 + (threadIdx.x >> 5);
  const int tilesPerBatch = Nn / 16;                   // 1250 (exact)
  const int b    = wave / tilesPerBatch;               // grid sized exactly: no partial waves
  const int tile = wave % tilesPerBatch;
  const int m    = lane & 15;
  const int hi   = lane >> 4;                          // 0: K+{0,1}, 1: K+{2,3}
  const int node = tile * 16 + m;

  // Three A fragments (x is [B,T,N]; t-stride = Nn)
  v2f a[3];
#pragma unroll
  for (int c = 0; c < 3; ++c) {
    const int t0 = 4 * c + 2 * hi;
    const float* xp = x + ((size_t)b * Tt + t0) * Nn + node;
    a[c].x = xp[0];
    a[c].y = xp[Nn];
  }

  float* hp = h + ((size_t)b * Nn + (size_t)tile * 16) * Ff;
#pragma unroll
  for (int j = 0; j < 4; ++j) {                        // four 16-wide feature tiles
    const int f0 = j * 16 + m;
    v8f d = {};
#pragma unroll
    for (int c = 0; c < 3; ++c) {
      const int t0 = 4 * c + 2 * hi;
      v2f bf;
      bf.x = w_gcn[(size_t)t0 * Ff + f0];
      bf.y = w_gcn[(size_t)(t0 + 1) * Ff + f0];
      d = __builtin_amdgcn_wmma_f32_16x16x4_f32(false, a[c], false, bf,
                                                (short)0, d, false, false);
    }
#pragma unroll
    for (int r = 0; r < 8; ++r) {
      const int row = r + 8 * hi;
      hp[(size_t)row * Ff + j * 16 + m] = d[r];
    }
  }
}

// ---------------------------------------------------------------------------
// Kernel 2: msg/scatter-add. 16 consecutive threads = one (b,e); each thread
// handles one float4 of the 64 features: coalesced 256B gather of h[b,src,:],
// then 4 global_atomic_add_f32 into L2-resident agg[b,dst,:].
// ---------------------------------------------------------------------------
__global__ __launch_bounds__(256) void edge_scatter(const float* __restrict__ h,
                                                    const float* __restrict__ ew,
                                                    const int* __restrict__ esrc,
                                                    const int* __restrict__ edst,
                                                    float* __restrict__ agg) {
  const long long idx   = (long long)blockIdx.x * blockDim.x + threadIdx.x;
  const long long total = (long long)Bb * Ee * (Ff / 4);
  if (idx >= total) return;
  const int fq      = (int)(idx & 15);
  const long long be = idx >> 4;
  const int e = (int)(be % Ee);
  const int b = (int)(be / Ee);

  const int   src = esrc[e];
  const int   dst = edst[e];
  const float w   = ew[e];

  const float4 v = *((const float4*)(h + ((size_t)b * Nn + src) * Ff) + fq);
  float* ap = agg + ((size_t)b * Nn + dst) * Ff + fq * 4;
  atomicAdd(ap + 0, v.x * w);
  atomicAdd(ap + 1, v.y * w);
  atomicAdd(ap + 2, v.z * w);
  atomicAdd(ap + 3, v.w * w);
}

// ---------------------------------------------------------------------------
// Kernel 3: out[b,n,t] = relu( relu(agg+b_gcn) @ w_dense + b_dense )
// One wave per (b, 16-node tile). K=64 -> 16 chunks of V_WMMA_F32_16X16X4_F32.
// ReLU+bias on A fused pre-WMMA; output tile is 16x16 with cols 12..15 zeroed
// in B and masked on store (T=12).
// ---------------------------------------------------------------------------
__global__ __launch_bounds__(256) void gcn_gemm2_wmma(const float* __restrict__ agg,
                                                      const float* __restrict__ b_gcn,
                                                      const float* __restrict__ w_dense,
                                                      const float* __restrict__ b_dense,
                                                      float* __restrict__ out) {
  const int lane = threadIdx.x & 31;
  const int wave = blockIdx.x * 8 + (threadIdx.x >> 5);
  const int tilesPerBatch = Nn / 16;
  const int b    = wave / tilesPerBatch;
  const int tile = wave % tilesPerBatch;
  const int m    = lane & 15;
  const int hi   = lane >> 4;

  const float* arow = agg + ((size_t)b * Nn + (size_t)tile * 16 + m) * Ff;
  const int col = m;                                   // output column for B/D
  const bool colOk = (col < Tt);

  v8f d = {};
#pragma unroll
  for (int c = 0; c < 16; ++c) {
    const int k0 = 4 * c + 2 * hi;
    v2f af, bf;
    af.x = fmaxf(arow[k0]     + b_gcn[k0],     0.0f);
    af.y = fmaxf(arow[k0 + 1] + b_gcn[k0 + 1], 0.0f);
    bf.x = colOk ? w_dense[(size_t)k0 * Tt + col]       : 0.0f;
    bf.y = colOk ? w_dense[(size_t)(k0 + 1) * Tt + col] : 0.0f;
    d = __builtin_amdgcn_wmma_f32_16x16x4_f32(false, af, false, bf,
                                              (short)0, d, false, false);
  }

  if (colOk) {
    const float bd = b_dense[col];
    float* op = out + ((size_t)b * Nn + (size_t)tile * 16) * Tt + col;
#pragma unroll
    for (int r = 0; r < 8; ++r) {
      const int row = r + 8 * hi;
      op[(size_t)row * Tt] = fmaxf(d[r] + bd, 0.0f);
    }
  }
}

// ---------------------------------------------------------------------------
extern "C" void kernel_launch(void* const* d_in, const int* in_sizes, int n_in,
                              void* d_out, int out_size, void* d_ws, size_t ws_size,
                              hipStream_t stream) {
  const float* x       = (const float*)d_in[0];  // [B,T,N]
  const float* ew      = (const float*)d_in[1];  // [E]
  const float* w_gcn   = (const float*)d_in[2];  // [T,F]
  const float* b_gcn   = (const float*)d_in[3];  // [F]
  const float* w_dense = (const float*)d_in[4];  // [F,T]
  const float* b_dense = (const float*)d_in[5];  // [T]
  const int*   esrc    = (const int*)d_in[6];    // [E]
  const int*   edst    = (const int*)d_in[7];    // [E]
  float*       out     = (float*)d_out;          // [B,N,T]

  // Workspace: h then agg, each B*N*F floats (40.96 MB each)
  float* h   = (float*)d_ws;
  float* agg = h + (size_t)Bb * Nn * Ff;

  hipMemsetAsync(agg, 0, (size_t)Bb * Nn * Ff * sizeof(float), stream);

  const int waves  = Bb * (Nn / 16);               // 10000 full waves
  const int blocks = waves / 8;                    // 1250 blocks of 8 waves
  gcn_gemm1_wmma<<<blocks, 256, 0, stream>>>(x, w_gcn, h);

  const long long tot = (long long)Bb * Ee * (Ff / 4);
  edge_scatter<<<(int)((tot + 255) / 256), 256, 0, stream>>>(h, ew, esrc, edst, agg);

  gcn_gemm2_wmma<<<blocks, 256, 0, stream>>>(agg, b_gcn, w_dense, b_dense, out);
}